// ActivationBuffer_89249420410972
// MI455X (gfx1250) — compile-verified
//
#include <hip/hip_runtime.h>
#include <hip/hip_bf16.h>

typedef float v4f __attribute__((ext_vector_type(4)));

#define SLOT_COPY (-1)
#define SLOT_ZERO (-2)

// ---------------------------------------------------------------------------
// Kernel 0: initialize slotmap (d_ws) to SLOT_COPY. D*M ints = 1 MB.
// ---------------------------------------------------------------------------
__global__ void ab_fill_kernel(int* __restrict__ slotmap, long long n) {
    long long i = (long long)blockIdx.x * blockDim.x + threadIdx.x;
    long long stride = (long long)gridDim.x * blockDim.x;
    for (; i < n; i += stride) slotmap[i] = SLOT_COPY;
}

// ---------------------------------------------------------------------------
// Kernel 1: per-row mask scan + scatter of slot -> source-token map.
//   grid = D blocks, block = 256 threads, each thread owns T/256 tokens.
//   Also emits new_n_valid / new_index (as float, per harness output dtype).
// Mask assumed raw numpy bool layout: 1 byte per element.
// ---------------------------------------------------------------------------
__global__ void ab_scan_scatter_kernel(const unsigned char* __restrict__ mask,
                                       const int* __restrict__ n_valid,
                                       const int* __restrict__ index,
                                       int* __restrict__ slotmap,
                                       float* __restrict__ out_nvalid,
                                       float* __restrict__ out_index,
                                       int T, int M) {
    const int d   = blockIdx.x;
    const int tid = threadIdx.x;
    const int NT  = blockDim.x;                 // 256
    const int per = (T + NT - 1) / NT;          // tokens per thread (16 for T=4096)
    const unsigned char* mrow = mask + (long long)d * T;

    __shared__ int s_cnt[256];
    __shared__ int s_total;

    // local count of kept tokens
    const int base = tid * per;
    int cnt = 0;
    for (int k = 0; k < per; ++k) {
        int t = base + k;
        if (t < T) cnt += (mrow[t] != 0);
    }

    // block-wide inclusive scan (Hillis-Steele) over per-thread counts
    s_cnt[tid] = cnt;
    __syncthreads();
    for (int off = 1; off < NT; off <<= 1) {
        int v = (tid >= off) ? s_cnt[tid - off] : 0;
        __syncthreads();
        s_cnt[tid] += v;
        __syncthreads();
    }
    const int incl = s_cnt[tid];
    const int excl = incl - cnt;
    if (tid == NT - 1) s_total = incl;
    __syncthreads();
    const int acc  = s_total;                   // kept tokens in row d
    const int idx0 = index[d];

    // scatter: kept token t -> slot (index + rank) % M, store global token id
    int rank = excl;
    int* smrow = slotmap + (long long)d * M;
    for (int k = 0; k < per; ++k) {
        int t = base + k;
        if (t < T && mrow[t] != 0) {
            int slot = idx0 + rank;
            if (slot >= M) slot -= M;           // rank < T < M: single wrap max
            smrow[slot] = d * T + t;            // global activation row id
            ++rank;
        }
    }

    if (tid == 0) {
        // offsets[0] == -1 case: reference zeroes slot (index-1)%M.
        // Cannot collide with kept slots since acc <= T < M.
        if (mrow[0] == 0) {
            int zslot = idx0 - 1;
            if (zslot < 0) zslot += M;
            smrow[zslot] = SLOT_ZERO;
        }
        int nv = n_valid[d] + acc;
        if (nv > M) nv = M;
        int ni = idx0 + acc;
        if (ni >= M) ni -= M;
        out_nvalid[d] = (float)nv;
        out_index[d]  = (float)ni;
    }
}

// ---------------------------------------------------------------------------
// Kernel 2: single-pass streaming emit of new_cache via the CDNA5 async
// global<->LDS engine (ASYNCcnt path, ISA 15.18 / async_tensor doc).
//   One block per (d, slot) row; 192 threads x 16 B = 3072 B = one row.
//   Per lane:  global_load_async_to_lds_b128  (HBM -> LDS, no VGPR data)
//              s_wait_asynccnt 0              (async load vs async store from
//                                              same LDS bytes is NOT ordered)
//              global_store_async_from_lds_b128 (LDS -> HBM)
//   Waves use disjoint LDS regions -> no barrier; trailing ASYNCcnt is
//   drained by S_ENDPGM's implicit wait-idle. Zero rows (<= D of them) use a
//   plain non-temporal store.
// ---------------------------------------------------------------------------
__global__ void ab_emit_kernel(const float* __restrict__ cache,
                               const float* __restrict__ acts,
                               const int* __restrict__ slotmap,
                               float* __restrict__ out,
                               int dim4 /* = DIM/4, <= 256 */) {
    __shared__ v4f stage[256];                 // 16 B per thread staging
    const long long row = blockIdx.x;          // row in [0, D*M)
    const int t = threadIdx.x;                 // [0, dim4)
    const int s = slotmap[row];                // uniform per block -> scalar

    float* dstrow = out + row * (long long)(dim4 * 4);

    if (s == SLOT_ZERO) {
        v4f z = {0.f, 0.f, 0.f, 0.f};
        __builtin_nontemporal_store(z, (v4f*)dstrow + t);
        return;
    }

    const float* srcrow = (s == SLOT_COPY)
        ? (cache + row * (long long)(dim4 * 4))
        : (acts + (long long)s * (dim4 * 4));

    unsigned long long ga = (unsigned long long)(const void*)((const v4f*)srcrow + t);
    unsigned long long gd = (unsigned long long)(void*)((v4f*)dstrow + t);
    // Flat LDS address low 32 bits == LDS byte offset (aperture spec).
    unsigned lds = (unsigned)(unsigned long long)(void*)&stage[t];

    asm volatile("global_load_async_to_lds_b128 %0, %1, off"
                 :: "v"(lds), "v"(ga) : "memory");
    asm volatile("s_wait_asynccnt 0x0" ::: "memory");
    asm volatile("global_store_async_from_lds_b128 %0, %1, off"
                 :: "v"(gd), "v"(lds) : "memory");
}

// ---------------------------------------------------------------------------
// Launcher
// in:  [0] cache f32 [D,M,DIM]  [1] activations f32 [D*T,DIM]
//      [2] mask bool [D*T]      [3] n_valid i32 [D]   [4] index i32 [D]
// out: new_cache f32 [D,M,DIM] ++ new_n_valid [D] ++ new_index [D] (as f32)
// ws:  slotmap int[D*M] (1 MB)
// ---------------------------------------------------------------------------
extern "C" void kernel_launch(void* const* d_in, const int* in_sizes, int n_in,
                              void* d_out, int out_size, void* d_ws, size_t ws_size,
                              hipStream_t stream) {
    const float*         cache   = (const float*)d_in[0];
    const float*         acts    = (const float*)d_in[1];
    const unsigned char* mask    = (const unsigned char*)d_in[2];
    const int*           n_valid = (const int*)d_in[3];
    const int*           index   = (const int*)d_in[4];

    const long long cacheN = in_sizes[0];           // D*M*DIM
    const long long actsN  = in_sizes[1];           // D*T*DIM
    const int       DT     = in_sizes[2];           // D*T
    const int       D      = in_sizes[3];
    const int       T      = DT / D;
    const int       DIM    = (int)(actsN / DT);
    const int       M      = (int)(cacheN / D / DIM);

    float* out        = (float*)d_out;
    float* out_nvalid = out + cacheN;
    float* out_index  = out_nvalid + D;
    int*   slotmap    = (int*)d_ws;

    const long long rows = (long long)D * M;        // 262144

    ab_fill_kernel<<<512, 256, 0, stream>>>(slotmap, rows);
    ab_scan_scatter_kernel<<<D, 256, 0, stream>>>(mask, n_valid, index, slotmap,
                                                  out_nvalid, out_index, T, M);
    ab_emit_kernel<<<(unsigned int)rows, DIM / 4, 0, stream>>>(cache, acts, slotmap,
                                                               out, DIM / 4);
}